// GNN_SAGE_74285754351848
// MI455X (gfx1250) — compile-verified
//
#include <hip/hip_runtime.h>
#include <hip/hip_bf16.h>

// ---------------------------------------------------------------------------
// GraphSAGE 2-layer forward for MI455X (gfx1250, wave32).
//   h   = relu( (segsum(x[src])*deg_inv) @ W1l + b1 + x @ W1r )
//   out = meanpool_graph( (segsum(h[src])*deg_inv) @ W2l + b2 + h @ W2r )
// Layer-2 reassociated: project to scalar first (y = h@W2l), then scatter
// scalars -> 128x less edge traffic. Layer-1 GEMM: fp32 WMMA 16x16x4 with
// block-level LDS staging of A tiles via GLOBAL_LOAD_ASYNC_TO_LDS_B128.
// ---------------------------------------------------------------------------

#define N_NODES  50000
#define N_EDGES  800000
#define DFEAT    128
#define N_GRAPHS 64

typedef __attribute__((ext_vector_type(2))) float v2f;
typedef __attribute__((ext_vector_type(8))) float v8f;
typedef int v4i_vs __attribute__((vector_size(16)));   // matches builtin param type

#define AS_GLOBAL __attribute__((address_space(1)))
#define AS_LDS    __attribute__((address_space(3)))

#if defined(__gfx1250__) && __has_builtin(__builtin_amdgcn_global_load_async_to_lds_b128)
#define HAVE_ASYNC_LDS 1
#else
#define HAVE_ASYNC_LDS 0
#endif

__device__ __forceinline__ void wait_async_lds() {
#if HAVE_ASYNC_LDS
#if __has_builtin(__builtin_amdgcn_s_wait_asynccnt)
    __builtin_amdgcn_s_wait_asynccnt(0);
#else
    asm volatile("s_wait_asynccnt 0" ::: "memory");
#endif
#endif
}

// ---------------- zero fill ----------------
__global__ void k_zero(float* __restrict__ p, int n) {
    int i = blockIdx.x * blockDim.x + threadIdx.x;
    if (i < n) p[i] = 0.0f;
}

// ---------------- degree count (f32 atomics into deg) ----------------
__global__ void k_deg(const long long* __restrict__ ei, float* __restrict__ deg) {
    int e = blockIdx.x * blockDim.x + threadIdx.x;
    if (e < N_EDGES) {
        int d = (int)ei[(long long)N_EDGES + e];   // dst row
        atomicAdd(&deg[d], 1.0f);
    }
}

// deg -> 1/max(deg,1) in place
__global__ void k_deginv(float* __restrict__ deg) {
    int i = blockIdx.x * blockDim.x + threadIdx.x;
    if (i < N_NODES) deg[i] = 1.0f / fmaxf(deg[i], 1.0f);
}

// ---------------- layer-1 scatter: agg[dst] += x[src]  (one wave / edge) ----
__global__ void k_scatter_vec(const long long* __restrict__ ei,
                              const float* __restrict__ x,
                              float* __restrict__ agg) {
    long long t = (long long)blockIdx.x * blockDim.x + threadIdx.x;
    int e    = (int)(t >> 5);
    int lane = (int)(t & 31);
    if (e >= N_EDGES) return;
    int s = (int)ei[e];
    int d = (int)ei[(long long)N_EDGES + e];
    const float4 v = ((const float4*)(x + (size_t)s * DFEAT))[lane];  // 16B/lane
    float* dst = agg + (size_t)d * DFEAT + lane * 4;
    atomicAdd(dst + 0, v.x);
    atomicAdd(dst + 1, v.y);
    atomicAdd(dst + 2, v.z);
    atomicAdd(dst + 3, v.w);
}

// ---------------- layer-1 GEMM via fp32 WMMA + LDS-staged A tiles ----------
// One 256-thread block (8 waves) per 16-row m-tile; wave w -> n-cols [16w,16w+16).
// h[m,n] = relu( agg[m,:]*dinv[m] @ W1l + x[m,:] @ W1r + b1[n] )
__global__ __launch_bounds__(256) void k_gemm1_wmma(
    const float* __restrict__ agg, const float* __restrict__ x,
    const float* __restrict__ deg_inv,
    const float* __restrict__ W1l, const float* __restrict__ b1,
    const float* __restrict__ W1r, float* __restrict__ h)
{
    __shared__ float sA[16 * DFEAT];   // raw agg tile (8 KB)
    __shared__ float sX[16 * DFEAT];   // x tile       (8 KB)

    const int m0   = blockIdx.x * 16;
    const int wave = threadIdx.x >> 5;
    const int lane = threadIdx.x & 31;

    // ---- stage 16x128 agg and x tiles (2048 floats each; 2 float4 / thread)
    {
        const float* gA = agg + (size_t)m0 * DFEAT;
        const float* gX = x   + (size_t)m0 * DFEAT;
        for (int it = 0; it < 2; ++it) {
            const int idx = threadIdx.x + it * 256;       // float4 index 0..511
#if HAVE_ASYNC_LDS
            __builtin_amdgcn_global_load_async_to_lds_b128(
                (AS_GLOBAL v4i_vs*)(gA + idx * 4),
                (AS_LDS v4i_vs*)(sA + idx * 4), 0, 0);
            __builtin_amdgcn_global_load_async_to_lds_b128(
                (AS_GLOBAL v4i_vs*)(gX + idx * 4),
                (AS_LDS v4i_vs*)(sX + idx * 4), 0, 0);
#else
            ((float4*)sA)[idx] = ((const float4*)gA)[idx];
            ((float4*)sX)[idx] = ((const float4*)gX)[idx];
#endif
        }
        wait_async_lds();
        __syncthreads();
    }

    const int n0    = wave << 4;                 // this wave's 16-col slice
    const int mr    = lane & 15;                 // A: lane%16 = row in tile
    const int ncol  = n0 + (lane & 15);          // B: lane%16 = N col
    const int khalf = (lane >> 4) << 1;          // lanes 16-31 -> K pair +2

    const float  dinv  = deg_inv[m0 + mr];
    const float* sArow = sA + mr * DFEAT;
    const float* sXrow = sX + mr * DFEAT;

    v8f c = {};
    // agg' @ W1l
    for (int k0 = 0; k0 < DFEAT; k0 += 4) {
        const int kb = k0 + khalf;
        v2f a, b;
        a.x = sArow[kb]     * dinv;              // ds_load_b64 + v_pk_mul
        a.y = sArow[kb + 1] * dinv;
        b.x = W1l[(size_t)kb       * DFEAT + ncol];
        b.y = W1l[(size_t)(kb + 1) * DFEAT + ncol];
        c = __builtin_amdgcn_wmma_f32_16x16x4_f32(false, a, false, b,
                                                  (short)0, c, false, false);
    }
    // x @ W1r  (same accumulator)
    for (int k0 = 0; k0 < DFEAT; k0 += 4) {
        const int kb = k0 + khalf;
        v2f a, b;
        a.x = sXrow[kb];
        a.y = sXrow[kb + 1];
        b.x = W1r[(size_t)kb       * DFEAT + ncol];
        b.y = W1r[(size_t)(kb + 1) * DFEAT + ncol];
        c = __builtin_amdgcn_wmma_f32_16x16x4_f32(false, a, false, b,
                                                  (short)0, c, false, false);
    }

    // C/D layout: VGPR j -> M = j (lanes 0-15) / j+8 (lanes 16-31), N = lane%16
    const float bias  = b1[ncol];
    const int   rbase = m0 + ((lane >> 4) << 3);
    for (int j = 0; j < 8; ++j) {
        float v = c[j] + bias;
        h[(size_t)(rbase + j) * DFEAT + ncol] = v > 0.0f ? v : 0.0f;
    }
}

// ---------------- per-node dual matvec: y = h@W2l, r = h@W2r (wave/node) ----
__global__ void k_matvec2(const float* __restrict__ h,
                          const float* __restrict__ W2l,
                          const float* __restrict__ W2r,
                          float* __restrict__ y, float* __restrict__ r) {
    long long t = (long long)blockIdx.x * blockDim.x + threadIdx.x;
    int node = (int)(t >> 5);
    int lane = (int)(t & 31);
    if (node >= N_NODES) return;
    const float4 hv = ((const float4*)(h + (size_t)node * DFEAT))[lane];
    const float4 wl = ((const float4*)W2l)[lane];
    const float4 wr = ((const float4*)W2r)[lane];
    float py = hv.x * wl.x + hv.y * wl.y + hv.z * wl.z + hv.w * wl.w;
    float pr = hv.x * wr.x + hv.y * wr.y + hv.z * wr.z + hv.w * wr.w;
    for (int off = 16; off; off >>= 1) {
        py += __shfl_xor(py, off, 32);
        pr += __shfl_xor(pr, off, 32);
    }
    if (lane == 0) { y[node] = py; r[node] = pr; }
}

// ---------------- layer-2 scalar scatter: agg2s[dst] += y[src] --------------
__global__ void k_scatter_scalar(const long long* __restrict__ ei,
                                 const float* __restrict__ y,
                                 float* __restrict__ agg2s) {
    int e = blockIdx.x * blockDim.x + threadIdx.x;
    if (e < N_EDGES) {
        int s = (int)ei[e];
        int d = (int)ei[(long long)N_EDGES + e];
        atomicAdd(&agg2s[d], y[s]);
    }
}

// ---------------- graph mean pool (accumulate) ----------------
__global__ void k_pool(const float* __restrict__ agg2s,
                       const float* __restrict__ deg_inv,
                       const float* __restrict__ r,
                       const float* __restrict__ b2,
                       const long long* __restrict__ batch,
                       float* __restrict__ gsum, float* __restrict__ gcnt) {
    int i = blockIdx.x * blockDim.x + threadIdx.x;
    if (i >= N_NODES) return;
    float v = agg2s[i] * deg_inv[i] + b2[0] + r[i];
    int g = (int)batch[i];
    atomicAdd(&gsum[g], v);
    atomicAdd(&gcnt[g], 1.0f);
}

__global__ void k_final(const float* __restrict__ gsum,
                        const float* __restrict__ gcnt,
                        float* __restrict__ out) {
    int g = threadIdx.x;
    if (g < N_GRAPHS) out[g] = gsum[g] / fmaxf(gcnt[g], 1.0f);
}

// ---------------------------------------------------------------------------
extern "C" void kernel_launch(void* const* d_in, const int* in_sizes, int n_in,
                              void* d_out, int out_size, void* d_ws, size_t ws_size,
                              hipStream_t stream) {
    (void)in_sizes; (void)n_in; (void)out_size; (void)ws_size;
    const float*     x    = (const float*)d_in[0];
    const long long* ei   = (const long long*)d_in[1];   // int64 [2, E]
    /* d_in[2] = edge_attr (unused by reference math) */
    const long long* batch= (const long long*)d_in[3];   // int64 [N]
    const float*     W1l  = (const float*)d_in[4];
    const float*     b1   = (const float*)d_in[5];
    const float*     W1r  = (const float*)d_in[6];
    const float*     W2l  = (const float*)d_in[7];
    const float*     b2   = (const float*)d_in[8];
    const float*     W2r  = (const float*)d_in[9];
    float* out = (float*)d_out;

    // workspace layout (floats)
    const size_t NAL = 50048;                         // N_NODES aligned up
    float* ws    = (float*)d_ws;
    float* deg   = ws;                                // [NAL]   (becomes deg_inv)
    float* agg   = deg + NAL;                         // [N*D]
    float* h     = agg + (size_t)N_NODES * DFEAT;     // [N*D]
    float* y     = h   + (size_t)N_NODES * DFEAT;     // [NAL]
    float* r     = y   + NAL;                         // [NAL]
    float* agg2s = r   + NAL;                         // [NAL]
    float* gsum  = agg2s + NAL;                       // [64]
    float* gcnt  = gsum + N_GRAPHS;                   // [64]

    const int B = 256;
    // zero: deg+agg contiguous block, and agg2s+gsum+gcnt contiguous block
    {
        int n1 = (int)(NAL + (size_t)N_NODES * DFEAT);
        k_zero<<<(n1 + B - 1) / B, B, 0, stream>>>(deg, n1);
        int n2 = (int)(NAL + 2 * N_GRAPHS);
        k_zero<<<(n2 + B - 1) / B, B, 0, stream>>>(agg2s, n2);
    }
    // degree + inverse
    k_deg<<<(N_EDGES + B - 1) / B, B, 0, stream>>>(ei, deg);
    k_deginv<<<(N_NODES + B - 1) / B, B, 0, stream>>>(deg);
    // layer-1 neighbor aggregation (wave per edge)
    {
        long long threads = (long long)N_EDGES * 32;
        k_scatter_vec<<<(int)((threads + B - 1) / B), B, 0, stream>>>(ei, x, agg);
    }
    // layer-1 fused dual-GEMM + bias + relu (WMMA fp32, LDS-staged A)
    k_gemm1_wmma<<<N_NODES / 16, 256, 0, stream>>>(agg, x, deg, W1l, b1, W1r, h);
    // layer-2 projections (wave per node)
    {
        long long threads = (long long)N_NODES * 32;
        k_matvec2<<<(int)((threads + B - 1) / B), B, 0, stream>>>(h, W2l, W2r, y, r);
    }
    // layer-2 scalar aggregation
    k_scatter_scalar<<<(N_EDGES + B - 1) / B, B, 0, stream>>>(ei, y, agg2s);
    // pooling
    k_pool<<<(N_NODES + B - 1) / B, B, 0, stream>>>(agg2s, deg, r, b2, batch, gsum, gcnt);
    k_final<<<1, 64, 0, stream>>>(gsum, gcnt, out);
}